// ChannelAttention_2_69544110456908
// MI455X (gfx1250) — compile-verified
//
#include <hip/hip_runtime.h>
#include <hip/hip_bf16.h>
#include <math.h>

// ---------------------------------------------------------------------------
// Types for WMMA f32 16x16x4 (CDNA5 / gfx1250, wave32)
// ---------------------------------------------------------------------------
typedef __attribute__((ext_vector_type(2))) float v2f;
typedef __attribute__((ext_vector_type(8))) float v8f;

// ---------------------------------------------------------------------------
// Problem constants (from the reference)
// ---------------------------------------------------------------------------
#define BB   8
#define CC   512
#define SS   256     // 16x16 pooled spatial
#define KK   2       // NUM_KERNELS
#define DD   100     // LATENT
#define MM   (KK*DD) // 200

// ---------------------------------------------------------------------------
// Kernel 1: fused adaptive-avg-pool(32x32) + 2x2 avg  ==  8x8 block mean
// x: [B,C,128,128] -> v: [B,C,16,16] (= [B,C,S])
// ---------------------------------------------------------------------------
__global__ void pool8x8_kernel(const float* __restrict__ x, float* __restrict__ v) {
    int idx = blockIdx.x * blockDim.x + threadIdx.x;      // over B*C*S = 1048576
    int s  = idx & 255;
    long bc = idx >> 8;
    int sy = s >> 4, sx = s & 15;
    const float* p = x + bc * (128L * 128L) + (long)sy * 8 * 128 + (long)sx * 8;
    float sum = 0.f;
#pragma unroll
    for (int r = 0; r < 8; ++r) {
        const float4* q = (const float4*)(p + (long)r * 128);
        float4 a = q[0];
        float4 b = q[1];
        sum += a.x + a.y + a.z + a.w + b.x + b.y + b.z + b.w;
    }
    v[idx] = sum * (1.f / 64.f);
}

// ---------------------------------------------------------------------------
// Batched fp32 GEMM via V_WMMA_F32_16X16X4_F32, fully templated so all
// addressing reduces to immediate-offset loads off fixed base pointers.
// One wave (32 threads) computes a 32x32 macro-tile as 2x2 WMMA tiles
// (4 accumulators), sharing A fragments across column-tiles and B fragments
// across row-tiles.
//
// Out-of-range A rows / B cols are CLAMPED (not masked): garbage only lands
// in D rows/cols that the guarded store never writes.
//
//   grid.z enumerates (b,k): b = z / ZK, k = z % ZK
//   TRANSB:     B element (kd,n) read from B[n*LDB + kd]   (B^T row-major)
//   TRANSSTORE: D element (m,n) written to C[n*LDC + m]
//   NACC:       compile-time inner accumulation slices (sum over kernels)
// KD must be a multiple of 4 (true for all uses: 100/200/256/512).
// All leading dims are even -> v2f loads are 8B-aligned.
// ---------------------------------------------------------------------------
template<int M, int N, int KD, int LDA, int LDB, int LDC,
         int TRANSB, int TRANSSTORE, int NACC, int ZK>
__global__ void gemm_wmma_f32(const float* __restrict__ A,
                              const float* __restrict__ B,
                              float* __restrict__ C,
                              long sAb, long sAk,
                              long sBb, long sBk,
                              long sCb, long sCk,
                              long sAacc, long sBacc) {
    int z = blockIdx.z;
    int b = (ZK == 1) ? z : (z / ZK);
    int k = (ZK == 1) ? 0 : (z % ZK);
    A += (long)b * sAb + (long)k * sAk;
    B += (long)b * sBb + (long)k * sBk;
    C += (long)b * sCb + (long)k * sCk;

    int m0 = blockIdx.y * 32;
    int n0 = blockIdx.x * 32;

    int lane = threadIdx.x & 31;
    int half = lane >> 4;       // 0: K even pair / D rows 0..7 ; 1: K odd pair / rows 8..15
    int l    = lane & 15;
    int kh   = 2 * half;        // per-lane K sub-offset (0 or 2)

    // Clamped, loop-invariant source indices (see note above).
    int rA0 = m0 + l;        if (rA0 > M - 1) rA0 = M - 1;
    int rA1 = m0 + 16 + l;   if (rA1 > M - 1) rA1 = M - 1;
    int cB0 = n0 + l;        if (cB0 > N - 1) cB0 = N - 1;
    int cB1 = n0 + 16 + l;   if (cB1 > N - 1) cB1 = N - 1;

    v8f acc00 = {}, acc01 = {}, acc10 = {}, acc11 = {};

#pragma unroll
    for (int a = 0; a < NACC; ++a) {
        // Fixed base pointers; inner loop indexes with compile-time offsets.
        const float* pa0 = A + (long)a * sAacc + (long)rA0 * LDA + kh;
        const float* pa1 = A + (long)a * sAacc + (long)rA1 * LDA + kh;
        const float* pb0;
        const float* pb1;
        if (TRANSB) {
            pb0 = B + (long)a * sBacc + (long)cB0 * LDB + kh;
            pb1 = B + (long)a * sBacc + (long)cB1 * LDB + kh;
        } else {
            pb0 = B + (long)a * sBacc + (long)kh * LDB + cB0;
            pb1 = B + (long)a * sBacc + (long)kh * LDB + cB1;
        }

#pragma unroll 4
        for (int k0 = 0; k0 < KD; k0 += 4) {
            v2f a0 = *(const v2f*)(pa0 + k0);
            v2f a1 = *(const v2f*)(pa1 + k0);
            v2f b0, b1;
            if (TRANSB) {
                b0 = *(const v2f*)(pb0 + k0);
                b1 = *(const v2f*)(pb1 + k0);
            } else {
                b0.x = pb0[k0 * LDB];
                b0.y = pb0[k0 * LDB + LDB];
                b1.x = pb1[k0 * LDB];
                b1.y = pb1[k0 * LDB + LDB];
            }
            acc00 = __builtin_amdgcn_wmma_f32_16x16x4_f32(
                false, a0, false, b0, (short)0, acc00, false, false);
            acc01 = __builtin_amdgcn_wmma_f32_16x16x4_f32(
                false, a0, false, b1, (short)0, acc01, false, false);
            acc10 = __builtin_amdgcn_wmma_f32_16x16x4_f32(
                false, a1, false, b0, (short)0, acc10, false, false);
            acc11 = __builtin_amdgcn_wmma_f32_16x16x4_f32(
                false, a1, false, b1, (short)0, acc11, false, false);
        }
    }

    // Store: accumulator VGPR r of a 16x16 tile -> row (tile_m0 + half*8 + r),
    // col (tile_n0 + l).
    int col0 = n0 + l;
    int col1 = n0 + 16 + l;
    int rowB = m0 + half * 8;
#pragma unroll
    for (int r = 0; r < 8; ++r) {
        int r0 = rowB + r;        // row-tile 0
        int r1 = r0 + 16;         // row-tile 1
        if (TRANSSTORE) {
            if (r0 < M && col0 < N) C[(long)col0 * LDC + r0] = acc00[r];
            if (r0 < M && col1 < N) C[(long)col1 * LDC + r0] = acc01[r];
            if (r1 < M && col0 < N) C[(long)col0 * LDC + r1] = acc10[r];
            if (r1 < M && col1 < N) C[(long)col1 * LDC + r1] = acc11[r];
        } else {
            if (r0 < M && col0 < N) C[(long)r0 * LDC + col0] = acc00[r];
            if (r0 < M && col1 < N) C[(long)r0 * LDC + col1] = acc01[r];
            if (r1 < M && col0 < N) C[(long)r1 * LDC + col0] = acc10[r];
            if (r1 < M && col1 < N) C[(long)r1 * LDC + col1] = acc11[r];
        }
    }
}

// ---------------------------------------------------------------------------
// Row softmax (in place). One block per row, n <= 512.
// ---------------------------------------------------------------------------
__global__ void softmax_rows_kernel(float* __restrict__ data, int n) {
    __shared__ float red[256];
    long row = blockIdx.x;
    float* p = data + row * (long)n;
    int t = threadIdx.x;

    float mx = -INFINITY;
    for (int i = t; i < n; i += 256) mx = fmaxf(mx, p[i]);
    red[t] = mx; __syncthreads();
    for (int s = 128; s > 0; s >>= 1) {
        if (t < s) red[t] = fmaxf(red[t], red[t + s]);
        __syncthreads();
    }
    mx = red[0]; __syncthreads();

    float sum = 0.f;
    for (int i = t; i < n; i += 256) {
        float e = __expf(p[i] - mx);
        p[i] = e;
        sum += e;
    }
    red[t] = sum; __syncthreads();
    for (int s = 128; s > 0; s >>= 1) {
        if (t < s) red[t] += red[t + s];
        __syncthreads();
    }
    float inv = 1.f / red[0];
    for (int i = t; i < n; i += 256) p[i] *= inv;
}

// ---------------------------------------------------------------------------
// invnorm[row] = 1 / (||z_row||_2 + 1e-6), rows of length n
// ---------------------------------------------------------------------------
__global__ void rownorm_inv_kernel(const float* __restrict__ z,
                                   float* __restrict__ invn, int n) {
    __shared__ float red[256];
    long row = blockIdx.x;
    const float* p = z + row * (long)n;
    int t = threadIdx.x;
    float s = 0.f;
    for (int i = t; i < n; i += 256) { float x = p[i]; s += x * x; }
    red[t] = s; __syncthreads();
    for (int k = 128; k > 0; k >>= 1) {
        if (t < k) red[t] += red[t + k];
        __syncthreads();
    }
    if (t == 0) invn[row] = 1.f / (sqrtf(red[0]) + 1e-6f);
}

// ---------------------------------------------------------------------------
// G = softmax_n( Gram[m,n] * invn[m] * invn[n] ), in place on Gram.
// grid = (Mdim, B), block = 256, Mdim = 200 <= 256 (single pass).
// ---------------------------------------------------------------------------
__global__ void g_scale_softmax_kernel(float* __restrict__ gram,
                                       const float* __restrict__ invn, int Mdim) {
    __shared__ float red[256];
    int b = blockIdx.y, m = blockIdx.x;
    float* p = gram + ((long)b * Mdim + m) * (long)Mdim;
    const float* inv = invn + (long)b * Mdim;
    int t = threadIdx.x;
    float sm = inv[m];

    float x = (t < Mdim) ? p[t] * sm * inv[t] : -INFINITY;
    red[t] = x; __syncthreads();
    for (int s = 128; s > 0; s >>= 1) {
        if (t < s) red[t] = fmaxf(red[t], red[t + s]);
        __syncthreads();
    }
    float mx = red[0]; __syncthreads();

    float e = (t < Mdim) ? __expf(x - mx) : 0.f;
    red[t] = e; __syncthreads();
    for (int s = 128; s > 0; s >>= 1) {
        if (t < s) red[t] += red[t + s];
        __syncthreads();
    }
    float invsum = 1.f / red[0];
    if (t < Mdim) p[t] = e * invsum;
}

// ---------------------------------------------------------------------------
// attn[b,c] = sigmoid( mean_s( v[b,c,s] + out2[b,c,s] ) )
// One block (256 threads) per (b,c); S = 256.
// ---------------------------------------------------------------------------
__global__ void finalize_kernel(const float* __restrict__ v,
                                const float* __restrict__ o,
                                float* __restrict__ out) {
    __shared__ float red[256];
    long bc = blockIdx.x;
    int t = threadIdx.x;
    float f = v[bc * 256 + t] + o[bc * 256 + t];
    red[t] = f; __syncthreads();
    for (int s = 128; s > 0; s >>= 1) {
        if (t < s) red[t] += red[t + s];
        __syncthreads();
    }
    if (t == 0) {
        float m = red[0] * (1.f / 256.f);
        out[bc] = 1.f / (1.f + __expf(-m));
    }
}

// ---------------------------------------------------------------------------
// kernel_launch
// Inputs: d_in[0]=x [8,512,128,128] f32, d_in[1]=psi_w [2,256,100] f32,
//         d_in[2]=phi_w [2,256,100] f32, d_in[3]=out_w [256,256] f32
// Output: d_out = attn [8,512] f32 (== [B,C,1,1])
// ---------------------------------------------------------------------------
extern "C" void kernel_launch(void* const* d_in, const int* in_sizes, int n_in,
                              void* d_out, int out_size, void* d_ws, size_t ws_size,
                              hipStream_t stream) {
    const float* x    = (const float*)d_in[0];
    const float* psiw = (const float*)d_in[1];
    const float* phiw = (const float*)d_in[2];
    const float* outw = (const float*)d_in[3];
    float* out = (float*)d_out;

    // Workspace layout (floats)
    float* ws   = (float*)d_ws;
    float* v    = ws;                 // [B,C,S]          1,048,576
    float* lpsi = v    + 1048576;     // [B,K,D,C]          819,200
    float* z    = lpsi + 819200;      // [B,M,S]            409,600
    float* gram = z    + 409600;      // [B,M,M]            320,000
    float* invn = gram + 320000;      // [B,M]                1,600
    float* zp   = invn + 1600;        // [B,M,S]            409,600
    float* lphi = zp   + 409600;      // [B,K,C,D]          819,200
    float* outm = lphi + 819200;      // [B,C,S]          1,048,576
    float* out2 = outm + 1048576;     // [B,C,S]          1,048,576

    // 1) pool: x -> v
    pool8x8_kernel<<<4096, 256, 0, stream>>>(x, v);

    // 2) psi logits: per (b,k): [C x S] @ [S x D] -> stored transposed [d,c]
    {
        dim3 g((DD + 31) / 32, CC / 32, BB * KK);
        gemm_wmma_f32<CC, DD, SS, SS, DD, CC, 0, 1, 1, KK>
            <<<g, 32, 0, stream>>>(v, psiw, lpsi,
            (long)CC * SS, 0, 0, (long)SS * DD,
            (long)KK * DD * CC, (long)DD * CC, 0, 0);
    }
    // softmax over C (rows of length 512): B*K*D = 1600 rows
    softmax_rows_kernel<<<BB * KK * DD, 256, 0, stream>>>(lpsi, CC);

    // 3) z = A_v2l @ v : per (b,k): [D x C] @ [C x S] -> [D x S]
    {
        dim3 g(SS / 32, (DD + 31) / 32, BB * KK);
        gemm_wmma_f32<DD, SS, CC, CC, SS, SS, 0, 0, 1, KK>
            <<<g, 32, 0, stream>>>(lpsi, v, z,
            (long)KK * DD * CC, (long)DD * CC,
            (long)CC * SS, 0,
            (long)MM * SS, (long)DD * SS, 0, 0);
    }

    // 4) row inverse norms of z  (B*M = 1600 rows of length 256)
    rownorm_inv_kernel<<<BB * MM, 256, 0, stream>>>(z, invn, SS);

    // 5) Gram = z @ z^T : per b: [M x S] @ [S x M]
    {
        dim3 g((MM + 31) / 32, (MM + 31) / 32, BB);
        gemm_wmma_f32<MM, MM, SS, SS, SS, MM, 1, 0, 1, 1>
            <<<g, 32, 0, stream>>>(z, z, gram,
            (long)MM * SS, 0, (long)MM * SS, 0,
            (long)MM * MM, 0, 0, 0);
    }

    // 6) G = softmax( scaled Gram )
    {
        dim3 g(MM, BB);
        g_scale_softmax_kernel<<<g, 256, 0, stream>>>(gram, invn, MM);
    }

    // 7) zp = G @ z : per b: [M x M] @ [M x S]
    {
        dim3 g(SS / 32, (MM + 31) / 32, BB);
        gemm_wmma_f32<MM, SS, MM, MM, SS, SS, 0, 0, 1, 1>
            <<<g, 32, 0, stream>>>(gram, z, zp,
            (long)MM * MM, 0, (long)MM * SS, 0,
            (long)MM * SS, 0, 0, 0);
    }

    // 8) phi logits: per (b,k): [C x S] @ [S x D] -> [c,d] (natural layout)
    {
        dim3 g((DD + 31) / 32, CC / 32, BB * KK);
        gemm_wmma_f32<CC, DD, SS, SS, DD, DD, 0, 0, 1, KK>
            <<<g, 32, 0, stream>>>(v, phiw, lphi,
            (long)CC * SS, 0, 0, (long)SS * DD,
            (long)KK * CC * DD, (long)CC * DD, 0, 0);
    }
    // softmax over D (rows of length 100): B*K*C = 8192 rows
    softmax_rows_kernel<<<BB * KK * CC, 256, 0, stream>>>(lphi, DD);

    // 9) outm = sum_k A_l2v_k @ zp_k : per b: [C x D] @ [D x S], 2 acc slices
    {
        dim3 g(SS / 32, CC / 32, BB);
        gemm_wmma_f32<CC, SS, DD, DD, SS, SS, 0, 0, KK, 1>
            <<<g, 32, 0, stream>>>(lphi, zp, outm,
            (long)KK * CC * DD, 0, (long)MM * SS, 0,
            (long)CC * SS, 0,
            (long)CC * DD, (long)DD * SS);
    }

    // 10) out2 = outm @ out_w : per b: [C x S] @ [S x S]
    {
        dim3 g(SS / 32, CC / 32, BB);
        gemm_wmma_f32<CC, SS, SS, SS, SS, SS, 0, 0, 1, 1>
            <<<g, 32, 0, stream>>>(outm, outw, out2,
            (long)CC * SS, 0, 0, 0,
            (long)CC * SS, 0, 0, 0);
    }

    // 11) attn = sigmoid(mean(v + out2))
    finalize_kernel<<<BB * CC, 256, 0, stream>>>(v, out2, out);

    (void)in_sizes; (void)n_in; (void)out_size; (void)ws_size;
}